// GIN_Encoder_41738492182625
// MI455X (gfx1250) — compile-verified
//
#include <hip/hip_runtime.h>
#include <math.h>
#include <stdint.h>

// ---------------------------------------------------------------------------
// GIN encoder for MI455X (gfx1250, wave32).
//   * fp32 WMMA (v_wmma_f32_16x16x4_f32) for all GEMMs -> exact-precision path
//   * weights held transposed in LDS (320KB/WGP), conflict-free ds_load_b64
//   * tile staging via GLOBAL_LOAD_ASYNC_TO_LDS_B128 (ASYNCcnt path, no VGPRs)
//   * segment_sum via native global_atomic_add_f32 (unsafeAtomicAdd)
//   * batchnorm stats: regs -> ds_add_f32 (block) -> 256 global atomics/block
// ---------------------------------------------------------------------------

#define NNODES  200000
#define NEDGES  600000
#define NGRAPHS 10000

static_assert(NNODES % 16 == 0, "full 16-row tiles");
static_assert(NGRAPHS % 16 == 0, "full 16-row tiles");

typedef __attribute__((ext_vector_type(2))) float v2f;
typedef __attribute__((ext_vector_type(8))) float v8f;

#define STAGE_PITCH 132   // 128 + 4 pad -> conflict-free ds_load_b64 fragments

static __device__ __forceinline__ v8f v8zero() {
    v8f z = {0.f, 0.f, 0.f, 0.f, 0.f, 0.f, 0.f, 0.f};
    return z;
}

static __device__ __forceinline__ unsigned lds_off(const void* p) {
    // generic shared pointer: low 32 bits == byte offset inside wave's LDS
    return (unsigned)(uintptr_t)p;
}

static __device__ __forceinline__ void async_ld_b128(unsigned lds, const void* g) {
    asm volatile("global_load_async_to_lds_b128 %0, %1, off"
                 :: "v"(lds), "v"((unsigned long long)(uintptr_t)g) : "memory");
}
static __device__ __forceinline__ void wait_async0() {
    asm volatile("s_wait_asynccnt 0" ::: "memory");
}
static __device__ __forceinline__ void wait_ds0() {
    asm volatile("s_wait_dscnt 0" ::: "memory");
}

// ---------------------------------------------------------------------------
__global__ void zero_kernel(float* __restrict__ p, int n) {
    int i = blockIdx.x * blockDim.x + threadIdx.x;
    if (i < n) p[i] = 0.0f;
}

// agg[dst[e], f] += h[src[e], f]   (segment_sum over edges)
__global__ void gather_kernel(const float* __restrict__ h,
                              const int* __restrict__ src,
                              const int* __restrict__ dst,
                              float* __restrict__ agg,
                              int F, int pin, int pagg) {
    unsigned idx = blockIdx.x * blockDim.x + threadIdx.x;
    unsigned total = (unsigned)NEDGES * (unsigned)F;
    if (idx >= total) return;
    int e = (int)(idx / (unsigned)F);
    int f = (int)(idx - (unsigned)e * (unsigned)F);
    unsafeAtomicAdd(&agg[dst[e] * pagg + f], h[src[e] * pin + f]);
}

// ---------------------------------------------------------------------------
// Fused GIN MLP:  z = h + agg ;  t = relu(z@W1+b1) ;  y = relu(t@W2+b2)
// writes y to zout and accumulates per-feature sum / sum^2 into stats[256].
// One wave owns a 16-row node tile; 8 column tiles of 16 -> 128 outputs.
// ---------------------------------------------------------------------------
template <int KDIM, int KPAD, bool L0>
__global__ __launch_bounds__(256) void gin_mlp_kernel(
    const float* __restrict__ hin, const float* __restrict__ agg,
    const float* __restrict__ W1, const float* __restrict__ b1,
    const float* __restrict__ W2, const float* __restrict__ b2,
    float* __restrict__ zout, float* __restrict__ stats)
{
    constexpr int W1P  = KPAD + 4;            // LDS pitch of W1^T
    constexpr int W2P  = 132;                 // LDS pitch of W2^T
    constexpr int PIN  = L0 ? 78 : 128;
    constexpr int PAGG = L0 ? 80 : 128;

    extern __shared__ char smem[];
    float* w1t    = (float*)smem;             // [128][W1P]   W1 transposed
    float* w2t    = w1t + 128 * W1P;          // [128][W2P]   W2 transposed
    float* bias   = w2t + 128 * W2P;          // [256] b1|b2
    float* bstat  = bias + 256;               // [256] block-local BN stats
    float* stageH = bstat + 256;              // [8][16][STAGE_PITCH]  h  (or z for L0)
    float* stageA = stageH + 8 * 16 * STAGE_PITCH; // [8][16][..] agg, then t

    const int tid = threadIdx.x;

    for (int i = tid; i < KDIM * 128; i += 256) {
        int k = i >> 7, n = i & 127;
        w1t[n * W1P + k] = W1[i];
    }
    if (KPAD > KDIM) {                        // zero-pad K to multiple of 4
        for (int i = tid; i < 128 * (KPAD - KDIM); i += 256) {
            int n = i / (KPAD - KDIM);
            int k = KDIM + (i - n * (KPAD - KDIM));
            w1t[n * W1P + k] = 0.0f;
        }
    }
    for (int i = tid; i < 128 * 128; i += 256) {
        int k = i >> 7, n = i & 127;
        w2t[n * W2P + k] = W2[i];
    }
    if (tid < 128) { bias[tid] = b1[tid]; bias[128 + tid] = b2[tid]; }
    bstat[tid & 255] = 0.0f;
    __syncthreads();

    const int wave  = tid >> 5, lane = tid & 31;
    const int row16 = lane & 15, hi = lane >> 4;   // hi: lane half (K-offset)
    float* myH = stageH + wave * 16 * STAGE_PITCH;
    float* myA = stageA + wave * 16 * STAGE_PITCH;
    const unsigned myHoff = lds_off(myH);
    const unsigned myAoff = lds_off(myA);

    float ssum[8], ssq[8];
#pragma unroll
    for (int i = 0; i < 8; ++i) { ssum[i] = 0.f; ssq[i] = 0.f; }

    const int numTiles = NNODES >> 4;
    for (int tile = blockIdx.x * 8 + wave; tile < numTiles; tile += gridDim.x * 8) {
        const int base = tile << 4;

        // ---- stage h / agg tiles into wave-private LDS --------------------
        if (L0) {
            // odd 78-float rows: synchronous staging of z = x + agg (padded)
            for (int idx = lane; idx < 16 * KPAD; idx += 32) {
                int r = idx / KPAD, c = idx - r * KPAD;
                float z = 0.f;
                if (c < KDIM)
                    z = hin[(base + r) * PIN + c] + agg[(base + r) * PAGG + c];
                myH[r * STAGE_PITCH + c] = z;
            }
            wait_ds0();
        } else {
            wait_ds0();   // previous tile's DS reads retired before overwrite
#pragma unroll
            for (int j = 0; j < 16; ++j) {
                unsigned lofs = (unsigned)(j * (STAGE_PITCH * 4)) + (unsigned)(lane << 4);
                async_ld_b128(myHoff + lofs, hin + (base + j) * 128 + (lane << 2));
                async_ld_b128(myAoff + lofs, agg + (base + j) * 128 + (lane << 2));
            }
            wait_async0();
        }

        // ---- GEMM 1: (16 x KPAD) @ (KPAD x 128) ---------------------------
        v8f acc[8];
#pragma unroll
        for (int ct = 0; ct < 8; ++ct) acc[ct] = v8zero();

#pragma unroll 4
        for (int ks = 0; ks < KPAD / 4; ++ks) {
            const int kb = ks * 4 + hi * 2;
            v2f a;
            if (L0) {
                a = *(const v2f*)(myH + row16 * STAGE_PITCH + kb);
            } else {
                v2f ah = *(const v2f*)(myH + row16 * STAGE_PITCH + kb);
                v2f aa = *(const v2f*)(myA + row16 * STAGE_PITCH + kb);
                a = ah + aa;                    // z = h + agg, fused in-loop
            }
#pragma unroll
            for (int ct = 0; ct < 8; ++ct) {
                v2f b = *(const v2f*)(w1t + (ct * 16 + row16) * W1P + kb);
                acc[ct] = __builtin_amdgcn_wmma_f32_16x16x4_f32(
                    false, a, false, b, (short)0, acc[ct], false, false);
            }
        }

        // ---- bias + relu, write t into stageA as A of GEMM 2 --------------
#pragma unroll
        for (int ct = 0; ct < 8; ++ct) {
            float bv = bias[ct * 16 + row16];
#pragma unroll
            for (int v = 0; v < 8; ++v) {
                float t = acc[ct][v] + bv;
                t = t > 0.f ? t : 0.f;
                myA[(v + 8 * hi) * STAGE_PITCH + ct * 16 + row16] = t;
            }
            acc[ct] = v8zero();
        }
        wait_ds0();

        // ---- GEMM 2: (16 x 128) @ (128 x 128) -----------------------------
#pragma unroll 4
        for (int ks = 0; ks < 32; ++ks) {
            const int kb = ks * 4 + hi * 2;
            v2f a = *(const v2f*)(myA + row16 * STAGE_PITCH + kb);
#pragma unroll
            for (int ct = 0; ct < 8; ++ct) {
                v2f b = *(const v2f*)(w2t + (ct * 16 + row16) * W2P + kb);
                acc[ct] = __builtin_amdgcn_wmma_f32_16x16x4_f32(
                    false, a, false, b, (short)0, acc[ct], false, false);
            }
        }

        // ---- bias + relu, store y, accumulate BN stats (full tiles) -------
#pragma unroll
        for (int ct = 0; ct < 8; ++ct) {
            float bv = bias[128 + ct * 16 + row16];
#pragma unroll
            for (int v = 0; v < 8; ++v) {
                int gr = base + v + 8 * hi;
                float y = acc[ct][v] + bv;
                y = y > 0.f ? y : 0.f;
                zout[gr * 128 + ct * 16 + row16] = y;
                ssum[ct] += y;
                ssq[ct]  += y * y;
            }
        }
    }

    // stats: regs -> LDS atomics -> 256 global atomics per block
#pragma unroll
    for (int ct = 0; ct < 8; ++ct) {
        int f = ct * 16 + row16;
        atomicAdd(&bstat[f], ssum[ct]);
        atomicAdd(&bstat[128 + f], ssq[ct]);
    }
    __syncthreads();
    unsafeAtomicAdd(&stats[tid & 255], bstat[tid & 255]);
}

// ---------------------------------------------------------------------------
__global__ void bn_finalize_kernel(const float* __restrict__ stats,
                                   const float* __restrict__ gamma,
                                   const float* __restrict__ beta,
                                   float* __restrict__ ss) {
    int f = threadIdx.x;
    if (f < 128) {
        const float invN = 1.0f / (float)NNODES;
        float mean = stats[f] * invN;
        float var  = stats[128 + f] * invN - mean * mean;
        var = var > 0.f ? var : 0.f;
        float sc = gamma[f] * rsqrtf(var + 1e-5f);
        ss[f]       = sc;
        ss[128 + f] = beta[f] - mean * sc;
    }
}

__global__ void bn_apply_kernel(const float* __restrict__ z,
                                const float* __restrict__ ss,
                                float* __restrict__ h) {
    int idx = blockIdx.x * blockDim.x + threadIdx.x;   // float4 index
    if (idx >= NNODES * 32) return;
    int f = (idx & 31) << 2;
    float4 v = ((const float4*)z)[idx];
    float4 r;
    r.x = ss[f + 0] * v.x + ss[128 + f + 0];
    r.y = ss[f + 1] * v.y + ss[128 + f + 1];
    r.z = ss[f + 2] * v.z + ss[128 + f + 2];
    r.w = ss[f + 3] * v.w + ss[128 + f + 3];
    ((float4*)h)[idx] = r;
}

__global__ void pool_kernel(const float* __restrict__ h,
                            const int* __restrict__ batch,
                            float* __restrict__ pooled) {
    unsigned idx = blockIdx.x * blockDim.x + threadIdx.x;
    if (idx >= (unsigned)NNODES * 128u) return;
    int i = (int)(idx >> 7), f = (int)(idx & 127u);
    unsafeAtomicAdd(&pooled[batch[i] * 128 + f], h[idx]);
}

// out = relu(pooled @ fcW + fcb) : (G x 128) @ (128 x 64) via fp32 WMMA
__global__ __launch_bounds__(256) void fc_kernel(const float* __restrict__ pooled,
                                                 const float* __restrict__ fcW,
                                                 const float* __restrict__ fcb,
                                                 float* __restrict__ out) {
    constexpr int WP = 132;
    extern __shared__ char smem[];
    float* wt    = (float*)smem;              // [64][WP]  fcW transposed
    float* bias  = wt + 64 * WP;              // [64]
    float* stage = bias + 64;                 // [8][16][STAGE_PITCH]

    const int tid = threadIdx.x;
    for (int i = tid; i < 128 * 64; i += 256) {
        int k = i >> 6, n = i & 63;
        wt[n * WP + k] = fcW[i];
    }
    if (tid < 64) bias[tid] = fcb[tid];
    __syncthreads();

    const int wave = tid >> 5, lane = tid & 31;
    const int row16 = lane & 15, hi = lane >> 4;
    float* mystage = stage + wave * 16 * STAGE_PITCH;
    const unsigned myoff = lds_off(mystage);

    const int numTiles = NGRAPHS >> 4;
    for (int tile = blockIdx.x * 8 + wave; tile < numTiles; tile += gridDim.x * 8) {
        const int base = tile << 4;
        wait_ds0();
#pragma unroll
        for (int j = 0; j < 16; ++j) {
            unsigned lofs = (unsigned)(j * (STAGE_PITCH * 4)) + (unsigned)(lane << 4);
            async_ld_b128(myoff + lofs, pooled + (base + j) * 128 + (lane << 2));
        }
        wait_async0();

        v8f acc[4];
#pragma unroll
        for (int ct = 0; ct < 4; ++ct) acc[ct] = v8zero();

#pragma unroll 4
        for (int ks = 0; ks < 32; ++ks) {
            const int kb = ks * 4 + hi * 2;
            v2f a = *(const v2f*)(mystage + row16 * STAGE_PITCH + kb);
#pragma unroll
            for (int ct = 0; ct < 4; ++ct) {
                v2f b = *(const v2f*)(wt + (ct * 16 + row16) * WP + kb);
                acc[ct] = __builtin_amdgcn_wmma_f32_16x16x4_f32(
                    false, a, false, b, (short)0, acc[ct], false, false);
            }
        }
#pragma unroll
        for (int ct = 0; ct < 4; ++ct) {
            float bv = bias[ct * 16 + row16];
#pragma unroll
            for (int v = 0; v < 8; ++v) {
                int gr = base + v + 8 * hi;
                float y = acc[ct][v] + bv;
                out[gr * 64 + ct * 16 + row16] = y > 0.f ? y : 0.f;
            }
        }
    }
}

// ---------------------------------------------------------------------------
static inline int cdiv(int a, int b) { return (a + b - 1) / b; }

extern "C" void kernel_launch(void* const* d_in, const int* in_sizes, int n_in,
                              void* d_out, int out_size, void* d_ws, size_t ws_size,
                              hipStream_t stream) {
    const float* x      = (const float*)d_in[0];
    const int*   ei     = (const int*)d_in[1];
    const int*   srcIdx = ei;
    const int*   dstIdx = ei + NEDGES;
    const int*   batch  = (const int*)d_in[2];
    const float* W1_0 = (const float*)d_in[3];
    const float* b1_0 = (const float*)d_in[4];
    const float* W2_0 = (const float*)d_in[5];
    const float* b2_0 = (const float*)d_in[6];
    const float* g_0  = (const float*)d_in[7];
    const float* be_0 = (const float*)d_in[8];
    const float* W1s  = (const float*)d_in[9];
    const float* b1s  = (const float*)d_in[10];
    const float* W2s  = (const float*)d_in[11];
    const float* b2s  = (const float*)d_in[12];
    const float* gs   = (const float*)d_in[13];
    const float* bes  = (const float*)d_in[14];
    const float* fcW  = (const float*)d_in[15];
    const float* fcb  = (const float*)d_in[16];
    float* out = (float*)d_out;

    // workspace layout (floats): h | agg | zbuf | stats[256] | ss[256] | pooled
    float* h      = (float*)d_ws;
    float* agg    = h    + (size_t)NNODES * 128;
    float* zbuf   = agg  + (size_t)NNODES * 128;
    float* stats  = zbuf + (size_t)NNODES * 128;
    float* ss     = stats + 256;
    float* pooled = ss + 256;

    const int T = 256;
    const size_t ldsL0   = (size_t)(128 * 84  + 128 * 132 + 512 + 2 * 8 * 16 * STAGE_PITCH) * 4;
    const size_t ldsMain = (size_t)(128 * 132 + 128 * 132 + 512 + 2 * 8 * 16 * STAGE_PITCH) * 4;
    const size_t ldsFc   = (size_t)(64 * 132 + 64 + 8 * 16 * STAGE_PITCH) * 4;

    // ---- layer 0 (78 -> 128) ----------------------------------------------
    zero_kernel<<<cdiv(NNODES * 128, T), T, 0, stream>>>(agg, NNODES * 128);
    zero_kernel<<<1, 256, 0, stream>>>(stats, 256);
    gather_kernel<<<cdiv(NEDGES * 78, T), T, 0, stream>>>(x, srcIdx, dstIdx, agg, 78, 78, 80);
    gin_mlp_kernel<78, 80, true><<<512, T, ldsL0, stream>>>(x, agg, W1_0, b1_0, W2_0, b2_0, zbuf, stats);
    bn_finalize_kernel<<<1, 128, 0, stream>>>(stats, g_0, be_0, ss);
    bn_apply_kernel<<<cdiv(NNODES * 32, T), T, 0, stream>>>(zbuf, ss, h);

    // ---- layers 1..4 (128 -> 128) -----------------------------------------
    for (int l = 0; l < 4; ++l) {
        zero_kernel<<<cdiv(NNODES * 128, T), T, 0, stream>>>(agg, NNODES * 128);
        zero_kernel<<<1, 256, 0, stream>>>(stats, 256);
        gather_kernel<<<cdiv(NEDGES * 128, T), T, 0, stream>>>(h, srcIdx, dstIdx, agg, 128, 128, 128);
        gin_mlp_kernel<128, 128, false><<<512, T, ldsMain, stream>>>(
            h, agg, W1s + l * 128 * 128, b1s + l * 128,
            W2s + l * 128 * 128, b2s + l * 128, zbuf, stats);
        bn_finalize_kernel<<<1, 128, 0, stream>>>(stats, gs + l * 128, bes + l * 128, ss);
        bn_apply_kernel<<<cdiv(NNODES * 32, T), T, 0, stream>>>(zbuf, ss, h);
    }

    // ---- pooling + FC ------------------------------------------------------
    zero_kernel<<<cdiv(NGRAPHS * 128, T), T, 0, stream>>>(pooled, NGRAPHS * 128);
    pool_kernel<<<cdiv(NNODES * 128, T), T, 0, stream>>>(h, batch, pooled);
    fc_kernel<<<80, T, ldsFc, stream>>>(pooled, fcW, fcb, out);

    (void)in_sizes; (void)n_in; (void)out_size; (void)ws_size;
}